// TopologyGNN_21251498181397
// MI455X (gfx1250) — compile-verified
//
#include <hip/hip_runtime.h>
#include <hip/hip_bf16.h>

#define N_AG 1024
#define FEAT_D 64
#define HDIM 128
#define TDIM 10
#define GAT_L 3

typedef __attribute__((ext_vector_type(2))) float v2f;
typedef __attribute__((ext_vector_type(8))) float v8f;

// ---------------------------------------------------------------------------
// Task encoder: task_h = relu(tf @ w1^T + b1) @ w2^T + b2  (tf is [1,10])
// Writes combined bias  bias2[n] = enc_b2[n] + task_h[n]  so the agent
// encoder's second GEMM can fuse the task broadcast into its bias epilogue.
// ---------------------------------------------------------------------------
__global__ void task_enc_kernel(const float* __restrict__ tf,
                                const float* __restrict__ w1,
                                const float* __restrict__ b1,
                                const float* __restrict__ w2,
                                const float* __restrict__ b2,
                                const float* __restrict__ enc_b2,
                                float* __restrict__ bias2) {
  __shared__ float t1[HDIM];
  int n = threadIdx.x;
  float s = b1[n];
#pragma unroll
  for (int k = 0; k < TDIM; ++k) s += tf[k] * w1[n * TDIM + k];
  t1[n] = fmaxf(s, 0.f);
  __syncthreads();
  float s2 = b2[n];
  for (int k = 0; k < HDIM; ++k) s2 += t1[k] * w2[n * HDIM + k];
  bias2[n] = s2 + enc_b2[n];
}

// ---------------------------------------------------------------------------
// fp32 WMMA GEMM:  C[M,N] = act( A[M,K] x opB + bias )
//   TRANSB=1: opB[k][n] = B[n*ldb + k]   (i.e. C = A @ B^T, weight layout)
//   TRANSB=0: opB[k][n] = B[k*ldb + n]   (i.e. C = A @ B)
// Block = 128 threads = 4 waves. Each wave owns a 16x32 C slab (two adjacent
// 16x16 tiles): one A fragment feeds two V_WMMA_F32_16X16X4_F32 per K-step,
// raising FLOP per loaded byte from 4 to 5.33 and halving A-side traffic.
// A fragment (16x4, ISA 7.12.2): lane = M row (lane&15); lane[4] selects the
// K pair {0,1} vs {2,3}; v2f = two consecutive K values.
// B fragment (4x16): lane = N col (lane&15); lane[4] selects K pair.
// C tile: vgpr v -> row m0 + v + 8*(lane>>4), col n0 + (lane&15).
// act: 0 = identity, 1 = relu, 2 = leaky_relu(0.2)
// ---------------------------------------------------------------------------
template <int TRANSB>
__global__ void wmma_gemm_f32(const float* __restrict__ A,
                              const float* __restrict__ B,
                              float* __restrict__ C,
                              int M, int N, int K,
                              int lda, int ldb, int ldc,
                              const float* __restrict__ bias, int act) {
  const int wave = threadIdx.x >> 5;
  const int lane = threadIdx.x & 31;
  const int hlf  = lane >> 4;                 // 0: K{0,1}, 1: K{2,3}
  const int m0 = (blockIdx.y * 4 + wave) << 4;
  const int n0 = blockIdx.x << 5;             // 32 columns per wave
  if (m0 >= M || n0 >= N) return;             // uniform per wave

  const int mr  = m0 + (lane & 15);
  const int nc0 = n0 + (lane & 15);
  const int nc1 = nc0 + 16;
  const float* __restrict__ Ap = A + (size_t)mr * lda + 2 * hlf;

  v8f acc0 = {};
  v8f acc1 = {};
#pragma unroll 4
  for (int k = 0; k < K; k += 4) {
    v2f a;
    a.x = Ap[k];
    a.y = Ap[k + 1];
    const int ka = k + 2 * hlf;
    v2f b0, b1;
    if (TRANSB) {
      b0.x = B[(size_t)nc0 * ldb + ka];
      b0.y = B[(size_t)nc0 * ldb + ka + 1];
      b1.x = B[(size_t)nc1 * ldb + ka];
      b1.y = B[(size_t)nc1 * ldb + ka + 1];
    } else {
      b0.x = B[(size_t)ka * ldb + nc0];
      b0.y = B[(size_t)(ka + 1) * ldb + nc0];
      b1.x = B[(size_t)ka * ldb + nc1];
      b1.y = B[(size_t)(ka + 1) * ldb + nc1];
    }
    acc0 = __builtin_amdgcn_wmma_f32_16x16x4_f32(
        false, a, false, b0, (short)0, acc0, false, false);
    acc1 = __builtin_amdgcn_wmma_f32_16x16x4_f32(
        false, a, false, b1, (short)0, acc1, false, false);
  }

  const float bv0 = bias ? bias[nc0] : 0.f;
  const float bv1 = bias ? bias[nc1] : 0.f;
#pragma unroll
  for (int v = 0; v < 8; ++v) {
    const int cm = m0 + v + 8 * hlf;
    float x0 = acc0[v] + bv0;
    float x1 = acc1[v] + bv1;
    if (act == 1) {
      x0 = fmaxf(x0, 0.f);
      x1 = fmaxf(x1, 0.f);
    } else if (act == 2) {
      x0 = (x0 > 0.f) ? x0 : 0.2f * x0;
      x1 = (x1 > 0.f) ? x1 : 0.2f * x1;
    }
    C[(size_t)cm * ldc + nc0] = x0;
    C[(size_t)cm * ldc + nc1] = x1;
  }
}

// ---------------------------------------------------------------------------
// f1[i] = Wh[i,:] . a[:H];  f2[i] = Wh[i,:] . a[H:]   (one block per row)
// ---------------------------------------------------------------------------
__global__ void fvec_kernel(const float* __restrict__ Wh,
                            const float* __restrict__ a,
                            float* __restrict__ f1, float* __restrict__ f2) {
  __shared__ float s1[HDIM], s2[HDIM];
  const int row = blockIdx.x, t = threadIdx.x;
  const float w = Wh[(size_t)row * HDIM + t];
  s1[t] = w * a[t];
  s2[t] = w * a[HDIM + t];
  __syncthreads();
  for (int off = 64; off > 0; off >>= 1) {
    if (t < off) { s1[t] += s1[t + off]; s2[t] += s2[t + off]; }
    __syncthreads();
  }
  if (t == 0) { f1[row] = s1[0]; f2[row] = s2[0]; }
}

// ---------------------------------------------------------------------------
// Row-wise masked softmax: att[i,j] = softmax_j( adj!=0 ? f1[i]+f2[j] : -1e9 )
// ---------------------------------------------------------------------------
__global__ void att_softmax_kernel(const float* __restrict__ adj,
                                   const float* __restrict__ f1,
                                   const float* __restrict__ f2,
                                   float* __restrict__ att) {
  __shared__ float red[256];
  const int row = blockIdx.x, t = threadIdx.x;
  const float f1r = f1[row];
  const float* __restrict__ arow = adj + (size_t)row * N_AG;
  float* __restrict__ orow = att + (size_t)row * N_AG;

  float m = -3.0e38f;
  for (int j = t; j < N_AG; j += 256) {
    const float e = (arow[j] == 0.f) ? -1e9f : (f1r + f2[j]);
    m = fmaxf(m, e);
  }
  red[t] = m; __syncthreads();
  for (int off = 128; off > 0; off >>= 1) {
    if (t < off) red[t] = fmaxf(red[t], red[t + off]);
    __syncthreads();
  }
  m = red[0]; __syncthreads();

  float sum = 0.f;
  for (int j = t; j < N_AG; j += 256) {
    const float e = (arow[j] == 0.f) ? -1e9f : (f1r + f2[j]);
    const float ex = __expf(e - m);
    orow[j] = ex;
    sum += ex;
  }
  red[t] = sum; __syncthreads();
  for (int off = 128; off > 0; off >>= 1) {
    if (t < off) red[t] += red[t + off];
    __syncthreads();
  }
  const float inv = 1.f / red[0];
  for (int j = t; j < N_AG; j += 256) orow[j] *= inv;
}

// ---------------------------------------------------------------------------
// Pairwise decoder: for i<j,
//   p = sigmoid( sum_h relu(Ad[i,h] + Bd[j,h] + b1[h]) * w2[h] + b2 )
//   out[i,j] = out[j,i] = p;  out[i,i] = 0.
// Blocks with bj < bi skip (covered by mirror writes).
// ---------------------------------------------------------------------------
__global__ void pairwise_kernel(const float* __restrict__ Ad,
                                const float* __restrict__ Bd,
                                const float* __restrict__ b1,
                                const float* __restrict__ w2,
                                const float* __restrict__ b2,
                                float* __restrict__ out) {
  const int bi = blockIdx.y, bj = blockIdx.x;
  if (bj < bi) return;
  __shared__ float As[16][HDIM + 1];
  __shared__ float Bs[16][HDIM + 1];
  __shared__ float b1s[HDIM], w2s[HDIM];
  const int t = threadIdx.y * 16 + threadIdx.x;
  for (int e = t; e < 16 * HDIM; e += 256) {
    const int r = e >> 7, c = e & 127;
    As[r][c] = Ad[(size_t)(bi * 16 + r) * HDIM + c];
    Bs[r][c] = Bd[(size_t)(bj * 16 + r) * HDIM + c];
  }
  if (t < HDIM) { b1s[t] = b1[t]; w2s[t] = w2[t]; }
  __syncthreads();

  const int ti = threadIdx.y, tj = threadIdx.x;
  const int i = bi * 16 + ti, j = bj * 16 + tj;
  if (i == j) { out[(size_t)i * N_AG + j] = 0.f; return; }
  if (i > j) return;  // only possible in diagonal blocks; mirror covers it

  float logit = b2[0];
#pragma unroll 8
  for (int hh = 0; hh < HDIM; ++hh) {
    float s = As[ti][hh] + Bs[tj][hh] + b1s[hh];
    s = fmaxf(s, 0.f);
    logit += s * w2s[hh];
  }
  const float p = 1.f / (1.f + __expf(-logit));
  out[(size_t)i * N_AG + j] = p;
  out[(size_t)j * N_AG + i] = p;
}

// ---------------------------------------------------------------------------
extern "C" void kernel_launch(void* const* d_in, const int* in_sizes, int n_in,
                              void* d_out, int out_size, void* d_ws, size_t ws_size,
                              hipStream_t stream) {
  const float* agent   = (const float*)d_in[0];
  const float* taskf   = (const float*)d_in[1];
  const float* adj     = (const float*)d_in[2];
  const float* enc_w1  = (const float*)d_in[3];
  const float* enc_b1  = (const float*)d_in[4];
  const float* enc_w2  = (const float*)d_in[5];
  const float* enc_b2  = (const float*)d_in[6];
  const float* task_w1 = (const float*)d_in[7];
  const float* task_b1 = (const float*)d_in[8];
  const float* task_w2 = (const float*)d_in[9];
  const float* task_b2 = (const float*)d_in[10];
  const float* gat_W   = (const float*)d_in[11];
  const float* gat_a   = (const float*)d_in[12];
  const float* dec_w1  = (const float*)d_in[13];
  const float* dec_b1  = (const float*)d_in[14];
  const float* dec_w2  = (const float*)d_in[15];
  const float* dec_b2  = (const float*)d_in[16];
  float* out = (float*)d_out;
  float* ws  = (float*)d_ws;

  // Workspace layout (floats)
  float* h     = ws;                       // 1024*128
  float* t1    = ws + 131072;              // 1024*128 (enc hidden)
  float* Wh    = ws + 262144;              // 1024*128
  float* bias2 = ws + 393216;              // 128
  float* f1    = ws + 393344;              // 1024
  float* f2    = ws + 394368;              // 1024
  float* att   = ws + 395392;              // 1024*1024
  float* Ad    = att;                      // reused after GAT stack
  float* Bd    = att + 131072;

  const dim3 gblk(128);
  const dim3 ggrd(HDIM / 32, N_AG / 64);   // 16x32 slab per wave, 4 waves/block

  // Task encoder -> combined bias
  task_enc_kernel<<<1, 128, 0, stream>>>(taskf, task_w1, task_b1,
                                         task_w2, task_b2, enc_b2, bias2);

  // Agent encoder: t1 = relu(X @ W1^T + b1);  h = t1 @ W2^T + (b2 + task_h)
  wmma_gemm_f32<1><<<ggrd, gblk, 0, stream>>>(agent, enc_w1, t1,
      N_AG, HDIM, FEAT_D, FEAT_D, FEAT_D, HDIM, enc_b1, 1);
  wmma_gemm_f32<1><<<ggrd, gblk, 0, stream>>>(t1, enc_w2, h,
      N_AG, HDIM, HDIM, HDIM, HDIM, HDIM, bias2, 0);

  // GAT layers
  for (int l = 0; l < GAT_L; ++l) {
    wmma_gemm_f32<1><<<ggrd, gblk, 0, stream>>>(h, gat_W + (size_t)l * HDIM * HDIM, Wh,
        N_AG, HDIM, HDIM, HDIM, HDIM, HDIM, nullptr, 0);
    fvec_kernel<<<N_AG, 128, 0, stream>>>(Wh, gat_a + l * 2 * HDIM, f1, f2);
    att_softmax_kernel<<<N_AG, 256, 0, stream>>>(adj, f1, f2, att);
    // h = leaky_relu(att @ Wh)
    wmma_gemm_f32<0><<<ggrd, gblk, 0, stream>>>(att, Wh, h,
        N_AG, HDIM, N_AG, N_AG, HDIM, HDIM, nullptr, 2);
  }

  // Decoder projections: Ad = h @ dec_w1[:, :H]^T;  Bd = h @ dec_w1[:, H:]^T
  wmma_gemm_f32<1><<<ggrd, gblk, 0, stream>>>(h, dec_w1, Ad,
      N_AG, HDIM, HDIM, HDIM, 2 * HDIM, HDIM, nullptr, 0);
  wmma_gemm_f32<1><<<ggrd, gblk, 0, stream>>>(h, dec_w1 + HDIM, Bd,
      N_AG, HDIM, HDIM, HDIM, 2 * HDIM, HDIM, nullptr, 0);

  // Pairwise decode + symmetrize
  pairwise_kernel<<<dim3(N_AG / 16, N_AG / 16), dim3(16, 16), 0, stream>>>(
      Ad, Bd, dec_b1, dec_w2, dec_b2, out);
}